// TargetOpinionPairRepresentation_63840393888082
// MI455X (gfx1250) — compile-verified
//
#include <hip/hip_runtime.h>
#include <stdint.h>

// Native vector types (avoid HIP's class-type float4 so __builtin_nontemporal_store works)
typedef float        f4  __attribute__((ext_vector_type(4)));
typedef unsigned int v4u __attribute__((ext_vector_type(4)));
typedef int          v8i __attribute__((ext_vector_type(8)));
typedef int          v4i __attribute__((ext_vector_type(4)));

#ifndef __has_builtin
#define __has_builtin(x) 0
#endif

#if defined(__gfx1250__) && __has_builtin(__builtin_amdgcn_tensor_load_to_lds) && \
    __has_builtin(__builtin_amdgcn_s_wait_tensorcnt)
#define USE_TDM 1
#else
#define USE_TDM 0
#endif

namespace {
constexpr int S   = 512;           // spans per batch
constexpr int D   = 768;           // span dim
constexpr int E   = 128;           // distance-embedding dim
constexpr int T   = 64;            // targets per batch
constexpr int O   = 64;            // opinions per batch
constexpr int NB  = 14;            // bucket bins
constexpr int ROW  = 2 * D + E;    // 1664 floats per output row
constexpr int D4   = D / 4;        // 192 float4
constexpr int E4   = E / 4;        // 32 float4
constexpr int ROW4 = ROW / 4;      // 416 float4
} // namespace

__device__ __forceinline__ int iabs_(int x) { return x < 0 ? -x : x; }
__device__ __forceinline__ int imin_(int x, int y) { return x < y ? x : y; }

#if USE_TDM
// 1-D contiguous f32 tile, global -> LDS, via the Tensor Data Mover.
// Descriptor bitfields per CDNA5 ISA (D# groups 0/1).
__device__ __forceinline__ void tdm_load_1d(uint32_t lds_addr, const void* gptr,
                                            uint32_t nelem /* f32 elements, < 65536 */) {
  const uint64_t ga = (uint64_t)(size_t)gptr;
  v4u g0;
  g0[0] = 1u;                                            // count=1 (valid), user mode
  g0[1] = lds_addr;                                      // lds_addr [63:32]
  g0[2] = (uint32_t)ga;                                  // global_addr[31:0]
  g0[3] = ((uint32_t)(ga >> 32) & 0x01FFFFFFu)           // global_addr[56:32]
          | (2u << 30);                                  // type=2 ("image")
  v8i g1;
  g1[0] = (int)(2u << 16);                               // data_size=2 (4B); wg_mask=0
  g1[1] = (int)((nelem & 0xFFFFu) << 16);                // tensor_dim0[15:0]
  g1[2] = (int)(((nelem >> 16) & 0xFFFFu) | (1u << 16)); // tensor_dim0[31:16], tensor_dim1=1
  g1[3] = (int)((nelem & 0xFFFFu) << 16);                // tile_dim0 = nelem
  g1[4] = 1;                                             // tile_dim1 = 1
  g1[5] = (int)nelem;                                    // tensor_dim0_stride[31:0]
  g1[6] = 0;                                             // stride hi / dim1_stride lo
  g1[7] = 0;
  const v4i z4 = (v4i){0, 0, 0, 0};
#if __clang_major__ >= 23
  const v8i z8 = (v8i){0, 0, 0, 0, 0, 0, 0, 0};
  __builtin_amdgcn_tensor_load_to_lds(g0, g1, z4, z4, z8, 0);
#else
  __builtin_amdgcn_tensor_load_to_lds(g0, g1, z4, z4, 0);
#endif
}
#endif // USE_TDM

__global__ __launch_bounds__(192)
void TargetOpinionPairRepresentation_kernel(const float* __restrict__ spans,
                                            const float* __restrict__ dist,
                                            const int*   __restrict__ sidx,
                                            const int*   __restrict__ tgt,
                                            const int*   __restrict__ opi,
                                            float*       __restrict__ out) {
  __shared__ f4 ldsT4[D4];       // this block's target span row (3 KB)
  __shared__ f4 ldsD4[NB * E4];  // full distance-embedding table (7 KB)

  const int bt  = blockIdx.x;    // b*T + t
  const int b   = bt / T;
  const int tid = threadIdx.x;   // 0..191

  const int ti = tgt[bt];
  const int ta = sidx[2 * ti];
  const int tb = sidx[2 * ti + 1];
  const f4* spanT = (const f4*)(spans + ((size_t)b * S + ti) * D);

#if USE_TDM
  if (tid < 32) {                // wave 0 only: issue both DMAs, drain TENSORcnt
    tdm_load_1d((uint32_t)(size_t)&ldsT4[0], spanT, (uint32_t)D);
    tdm_load_1d((uint32_t)(size_t)&ldsD4[0], dist, (uint32_t)(NB * E));
    __builtin_amdgcn_s_wait_tensorcnt(0);
  }
  __syncthreads();
#else
  for (int i = tid; i < D4; i += 192)      ldsT4[i] = spanT[i];
  for (int i = tid; i < NB * E4; i += 192) ldsD4[i] = ((const f4*)dist)[i];
  __syncthreads();
#endif

  f4* outBT = (f4*)(out + (size_t)bt * O * ROW);
  const int bO = b * O;

  for (int o = 0; o < O; ++o) {
    const int oi = opi[bO + o];
    const int c  = sidx[2 * oi];
    const int d  = sidx[2 * oi + 1];
    const int w  = imin_(iabs_(tb - c), iabs_(ta - d));
    // searchsorted(BINS, w, 'right') - 1  ==  count of nonzero bins <= w  (w >= 0 always)
    const int bucket = (w >= 1) + (w >= 2) + (w >= 3) + (w >= 4) + (w >= 5) +
                       (w >= 7) + (w >= 8) + (w >= 15) + (w >= 16) +
                       (w >= 31) + (w >= 32) + (w >= 63) + (w >= 64);

    const f4* spanO = (const f4*)(spans + ((size_t)b * S + oi) * D);
    f4* orow = outBT + (size_t)o * ROW4;

    // Prefetch next opinion's span row (48 lanes x 64B = full 3 KB row) into L2/WGP$.
    if (o + 1 < O && tid < 48) {
      const int oin = opi[bO + o + 1];
      __builtin_prefetch(spans + ((size_t)b * S + oin) * D + tid * 16, 0, 3);
    }

    // 1664 floats = 192 + 192 + 32 float4: exactly one b128 NT store per thread
    // for the two span segments, lanes 0..31 also emit the distance segment.
    __builtin_nontemporal_store(ldsT4[tid], orow + tid);
    __builtin_nontemporal_store(spanO[tid], orow + D4 + tid);
    if (tid < E4)
      __builtin_nontemporal_store(ldsD4[bucket * E4 + tid], orow + 2 * D4 + tid);
  }
}

extern "C" void kernel_launch(void* const* d_in, const int* in_sizes, int n_in,
                              void* d_out, int out_size, void* d_ws, size_t ws_size,
                              hipStream_t stream) {
  const float* spans = (const float*)d_in[0];  // [B,S,D] f32
  const float* dist  = (const float*)d_in[1];  // [14,E]  f32
  const int*   sidx  = (const int*)d_in[2];    // [S,2]   i32
  const int*   tgt   = (const int*)d_in[3];    // [B,T]   i32
  const int*   opi   = (const int*)d_in[4];    // [B,O]   i32
  float*       out   = (float*)d_out;          // [B,T*O,2D+E] f32

  const int nBT = in_sizes[3];                 // B*T blocks, one per (batch, target)
  TargetOpinionPairRepresentation_kernel<<<dim3(nBT), dim3(192), 0, stream>>>(
      spans, dist, sidx, tgt, opi, out);
}